// GCNNodeEncoder_43293270344033
// MI455X (gfx1250) — compile-verified
//
#include <hip/hip_runtime.h>
#include <math.h>

// ---------------------------------------------------------------------------
// GCN node encoder for MI455X (gfx1250, wave32).
// GEMMs use V_WMMA_F32_16X16X4_F32 (exact fp32 matrix pipe).
// ---------------------------------------------------------------------------

typedef __attribute__((ext_vector_type(2))) float v2f;
typedef __attribute__((ext_vector_type(8))) float v8f;

#define DIM 128
#define BN_EPS 1e-5f

// ---------------- utility -------------------------------------------------

__global__ void zero_f32_kernel(float* __restrict__ p, int n) {
  int i = blockIdx.x * blockDim.x + threadIdx.x;
  if (i < n) p[i] = 0.0f;
}

// degree counting: deg_s / deg_d are int accumulators (zeroed beforehand)
__global__ void degree_kernel(const int* __restrict__ src, const int* __restrict__ dst,
                              int* __restrict__ deg_s, int* __restrict__ deg_d, int E) {
  int e = blockIdx.x * blockDim.x + threadIdx.x;
  if (e < E) {
    atomicAdd(&deg_s[src[e]], 1);
    atomicAdd(&deg_d[dst[e]], 1);
  }
}

// 1/sqrt(deg) with deg==0 -> 0  (matches reference _deg_inv_sqrt)
__global__ void norm_kernel(const int* __restrict__ deg_s, const int* __restrict__ deg_d,
                            float* __restrict__ onorm, float* __restrict__ inorm, int N) {
  int i = blockIdx.x * blockDim.x + threadIdx.x;
  if (i < N) {
    int ds = deg_s[i];
    int dd = deg_d[i];
    onorm[i] = (ds > 0) ? 1.0f / sqrtf((float)ds) : 0.0f;
    inorm[i] = (dd > 0) ? 1.0f / sqrtf((float)dd) : 0.0f;
  }
}

// x[i,:] = emb_table[node_ids[i],:]   (float4 per thread)
__global__ void gather_kernel(const int* __restrict__ ids, const float* __restrict__ emb,
                              float* __restrict__ x, int N) {
  int t = blockIdx.x * blockDim.x + threadIdx.x;
  int total = N * (DIM / 4);
  if (t >= total) return;
  int i = t / (DIM / 4);
  int c = (t % (DIM / 4)) * 4;
  const float4 v = *reinterpret_cast<const float4*>(emb + (size_t)ids[i] * DIM + c);
  *reinterpret_cast<float4*>(x + (size_t)i * DIM + c) = v;
}

// agg[dst,:] += x[src,:] * out_norm[src]   (one edge x 4 features per thread)
__global__ void scatter_kernel(const int* __restrict__ src, const int* __restrict__ dst,
                               const float* __restrict__ x, const float* __restrict__ onorm,
                               float* __restrict__ agg, int E) {
  long long t = (long long)blockIdx.x * blockDim.x + threadIdx.x;
  int e = (int)(t >> 5);
  if (e >= E) return;
  int c = ((int)t & 31) * 4;
  int s = src[e];
  int d = dst[e];
  float ns = onorm[s];
  const float4 v = *reinterpret_cast<const float4*>(x + (size_t)s * DIM + c);
  float* ap = agg + (size_t)d * DIM + c;
  atomicAdd(ap + 0, v.x * ns);
  atomicAdd(ap + 1, v.y * ns);
  atomicAdd(ap + 2, v.z * ns);
  atomicAdd(ap + 3, v.w * ns);
}

// ---------------- fused dual-GEMM + bias + ReLU + residual + BN stats -----
//
// h = relu((agg * in_norm) @ W + b) + relu(x @ Rw + Rb)
// also accumulates per-column sum / sum^2 for batch-norm.
//
// One wave computes a 16-row stripe x all 128 columns (8 col tiles).
// Per k-step (K=4): two v_wmma_f32_16x16x4_f32 (the two GEMMs).
// Fragment layouts per CDNA5 ISA 7.12.2:
//   A vgpr j : A[row0 + (lane&15), k0 + j + 2*(lane>>4)]
//   B vgpr j : B[k0 + j + 2*(lane>>4), col0 + (lane&15)]
//   C vgpr j : C[row0 + j + 8*(lane>>4), col0 + (lane&15)]
__global__ void __launch_bounds__(256) gemm_fused_kernel(
    const float* __restrict__ agg, const float* __restrict__ xin,
    const float* __restrict__ inorm,
    const float* __restrict__ W, const float* __restrict__ bias,
    const float* __restrict__ Rw, const float* __restrict__ Rb,
    float* __restrict__ h, float* __restrict__ csum, float* __restrict__ csumsq,
    int N) {
  const int wave = threadIdx.x >> 5;
  const int lane = threadIdx.x & 31;
  const int rowTile = blockIdx.x * 8 + wave;
  const int row0 = rowTile * 16;
  if (row0 >= N) return;  // wave-uniform guard: EXEC stays all-1s for WMMA

  const int r = lane & 15;
  const int half = lane >> 4;
  const int row = row0 + r;
  const float innr = inorm[row];
  const float* aggRow = agg + (size_t)row * DIM;
  const float* xRow = xin + (size_t)row * DIM;

  for (int ct = 0; ct < 8; ++ct) {
    const int col0 = ct * 16;
    v8f acc1 = {0.f, 0.f, 0.f, 0.f, 0.f, 0.f, 0.f, 0.f};
    v8f acc2 = {0.f, 0.f, 0.f, 0.f, 0.f, 0.f, 0.f, 0.f};
    const float* wc = W + col0 + r;
    const float* rc = Rw + col0 + r;
#pragma unroll 8
    for (int k0 = 0; k0 < DIM; k0 += 4) {
      const int kk = k0 + 2 * half;
      // A fragments: contiguous float2 per lane
      const float2 t1 = *reinterpret_cast<const float2*>(aggRow + kk);
      const float2 t2 = *reinterpret_cast<const float2*>(xRow + kk);
      v2f a1, a2, b1, b2;
      a1.x = t1.x * innr;  // fold D^{-1/2} (dst side) into A
      a1.y = t1.y * innr;
      a2.x = t2.x;
      a2.y = t2.y;
      // B fragments: column loads (coalesced across lanes)
      b1.x = wc[(size_t)kk * DIM];
      b1.y = wc[(size_t)(kk + 1) * DIM];
      b2.x = rc[(size_t)kk * DIM];
      b2.y = rc[(size_t)(kk + 1) * DIM];
      acc1 = __builtin_amdgcn_wmma_f32_16x16x4_f32(false, a1, false, b1,
                                                   (short)0, acc1, false, false);
      acc2 = __builtin_amdgcn_wmma_f32_16x16x4_f32(false, a2, false, b2,
                                                   (short)0, acc2, false, false);
    }
    const float bb = bias[col0 + r];
    const float rb = Rb[col0 + r];
    float s = 0.0f, s2 = 0.0f;
#pragma unroll
    for (int j = 0; j < 8; ++j) {
      float v = fmaxf(acc1[j] + bb, 0.0f) + fmaxf(acc2[j] + rb, 0.0f);
      h[(size_t)(row0 + j + 8 * half) * DIM + col0 + r] = v;
      s += v;
      s2 += v * v;
    }
    atomicAdd(&csum[col0 + r], s);
    atomicAdd(&csumsq[col0 + r], s2);
  }
}

// fold stats into per-column affine: scale = gamma/sqrt(var+eps); shift = beta - mu*scale
__global__ void bnprep_kernel(const float* __restrict__ csum, const float* __restrict__ csumsq,
                              const float* __restrict__ gamma, const float* __restrict__ beta,
                              float* __restrict__ scale, float* __restrict__ shift, float invN) {
  int c = threadIdx.x;  // launched with DIM threads
  float mu = csum[c] * invN;
  float var = csumsq[c] * invN - mu * mu;  // population variance (ddof=0)
  float sc = gamma[c] / sqrtf(var + BN_EPS);
  scale[c] = sc;
  shift[c] = beta[c] - mu * sc;
}

__global__ void bnapply_kernel(const float* __restrict__ h, const float* __restrict__ scale,
                               const float* __restrict__ shift, float* __restrict__ out,
                               int total) {
  int i = blockIdx.x * blockDim.x + threadIdx.x;
  if (i < total) {
    int c = i & (DIM - 1);
    out[i] = h[i] * scale[c] + shift[c];
  }
}

// ---------------- host-side orchestration ---------------------------------

static inline int cdiv(int a, int b) { return (a + b - 1) / b; }

extern "C" void kernel_launch(void* const* d_in, const int* in_sizes, int n_in,
                              void* d_out, int out_size, void* d_ws, size_t ws_size,
                              hipStream_t stream) {
  const int* node_ids = (const int*)d_in[0];
  const int* src = (const int*)d_in[1];
  const int* dst = (const int*)d_in[2];
  const float* emb = (const float*)d_in[3];
  const float* Ws = (const float*)d_in[4];
  const float* bs = (const float*)d_in[5];
  const float* Rws = (const float*)d_in[6];
  const float* Rbs = (const float*)d_in[7];
  const float* gammas = (const float*)d_in[8];
  const float* betas = (const float*)d_in[9];

  const int N = in_sizes[0];
  const int E = in_sizes[1];
  const int L = in_sizes[4] / (DIM * DIM);  // 3

  // workspace layout (floats)
  float* wsf = (float*)d_ws;
  size_t off = 0;
  float* onorm = wsf + off; off += (size_t)N;
  float* inorm = wsf + off; off += (size_t)N;
  float* x    = wsf + off;  off += (size_t)N * DIM;
  float* agg  = wsf + off;  off += (size_t)N * DIM;
  float* h    = wsf + off;  off += (size_t)N * DIM;
  float* csum   = wsf + off; off += DIM;
  float* csumsq = wsf + off; off += DIM;
  float* scale  = wsf + off; off += DIM;
  float* shift  = wsf + off; off += DIM;
  // degree counters alias the h buffer (h only written after degrees consumed)
  int* deg_s = (int*)h;
  int* deg_d = deg_s + N;

  const int B = 256;

  // 1) degrees -> normalizers
  zero_f32_kernel<<<cdiv(2 * N, B), B, 0, stream>>>((float*)deg_s, 2 * N);
  degree_kernel<<<cdiv(E, B), B, 0, stream>>>(src, dst, deg_s, deg_d, E);
  norm_kernel<<<cdiv(N, B), B, 0, stream>>>(deg_s, deg_d, onorm, inorm, N);

  // 2) embedding gather
  gather_kernel<<<cdiv(N * (DIM / 4), B), B, 0, stream>>>(node_ids, emb, x, N);

  // 3) layers
  const int ND = N * DIM;
  const int rowTiles = cdiv(N, 16);
  for (int l = 0; l < L; ++l) {
    zero_f32_kernel<<<cdiv(ND, B), B, 0, stream>>>(agg, ND);
    zero_f32_kernel<<<1, B, 0, stream>>>(csum, 2 * DIM);  // csum+csumsq contiguous

    scatter_kernel<<<cdiv(E * 32, B), B, 0, stream>>>(src, dst, x, onorm, agg, E);

    gemm_fused_kernel<<<cdiv(rowTiles, 8), B, 0, stream>>>(
        agg, x, inorm,
        Ws + (size_t)l * DIM * DIM, bs + (size_t)l * DIM,
        Rws + (size_t)l * DIM * DIM, Rbs + (size_t)l * DIM,
        h, csum, csumsq, N);

    bnprep_kernel<<<1, DIM, 0, stream>>>(csum, csumsq, gammas + (size_t)l * DIM,
                                         betas + (size_t)l * DIM, scale, shift,
                                         1.0f / (float)N);

    float* dest = (l == L - 1) ? (float*)d_out : x;
    bnapply_kernel<<<cdiv(ND, B), B, 0, stream>>>(h, scale, shift, dest, ND);
  }
}